// PixelateDegradation_1133871366348
// MI455X (gfx1250) — compile-verified
//
#include <hip/hip_runtime.h>
#include <cmath>

// Problem constants (match reference)
#define IMG     256
#define TSTEPS  20
#define BATCH   64
#define CH      3
#define MAT_EL  (IMG * IMG)          // 65536 floats per 256x256 matrix

typedef __attribute__((ext_vector_type(2))) float        v2f;
typedef __attribute__((ext_vector_type(8))) float        v8f;
typedef __attribute__((ext_vector_type(4))) unsigned int v4u;
typedef __attribute__((ext_vector_type(4))) int          v4i;
typedef __attribute__((ext_vector_type(8))) int          v8i;

// ---------------------------------------------------------------------------
// LDS tile geometry (floats).  Row strides are multiples of 4 floats (16-byte
// aligned DMA chunks) and chosen so the 16 lanes of each half-wave hit 16
// distinct banks on fragment reads.
//   A  tile: 128 rows x 16 k, stride 20  (80 B rows;  20*l mod 64 distinct)
//   B  tile (NN, Bs[k][n]):  16 x 64, stride 72 (k / k+2 rows land in
//                            disjoint 16-bank windows: shift 8/row)
//   B  tile (NT, Bts[n][k]): 64 x 16, stride 20 (fragment = contiguous float2)
// TDM descriptors reproduce these strides with pad_interval/pad_amount.
// ---------------------------------------------------------------------------
#define KC         16
#define A_STRIDE   20
#define A_TILE_F   (128 * A_STRIDE)              // 2560
#define BNN_STRIDE 72
#define BNT_STRIDE 20
#define B_TILE_F   1280                          // max(16*72, 64*20)
#define BUF_F      (A_TILE_F + B_TILE_F)         // 3840 floats / buffer

// 0-based LDS byte offset of a generic pointer to __shared__ memory
__device__ __forceinline__ unsigned lds_off_of(const void* p) {
    return (unsigned)(uintptr_t)(__attribute__((address_space(3))) const void*)p;
}

// per-lane async DMA: 16 bytes global -> LDS (ASYNCcnt)
__device__ __forceinline__ void async_copy16(unsigned lds_byte_off,
                                             const float* gsrc) {
    asm volatile("global_load_async_to_lds_b128 %0, %1, off"
                 :: "v"(lds_byte_off), "v"(gsrc)
                 : "memory");
}

__device__ __forceinline__ void wait_async0() {
    asm volatile("s_wait_asynccnt 0" ::: "memory");
}

// ---------------------------------------------------------------------------
// Tensor Data Mover: one descriptor-driven DMA per tile (TENSORcnt).
// 2D tile, 4-byte elements, row padding in LDS per D# pad fields.
//   pad_interval encoding: value v -> pad every 2^(v+1) DWORDs
//   pad_amount  encoding: value v -> insert (v+1) DWORDs
// ---------------------------------------------------------------------------
__device__ __forceinline__ void tdm_load_2d(unsigned lds_byte,
                                            const void* gptr,
                                            unsigned tile_d0,   // elems / row
                                            unsigned tile_d1,   // rows
                                            unsigned stride0,   // elems between rows
                                            unsigned pad_iv,    // encoded
                                            unsigned pad_amt)   // encoded
{
    unsigned long long ga = (unsigned long long)(uintptr_t)gptr;
    v4u g0;
    g0.x = 1u;                                        // count=1 valid descriptor
    g0.y = lds_byte;                                  // lds_addr
    g0.z = (unsigned)(ga & 0xFFFFFFFFu);              // global_addr[31:0]
    g0.w = (unsigned)((ga >> 32) & 0x01FFFFFFu)       // global_addr[56:32]
         | (2u << 30);                                // type = 2 ("image")
    v8i g1;
    g1[0] = (int)((2u << 16)                          // data_size = 4 B
                | (1u << 20)                          // pad_enable
                | (pad_iv << 22) | (pad_amt << 25));
    g1[1] = (int)((tile_d0 & 0xFFFFu) << 16);         // tensor_dim0[15:0]
    g1[2] = (int)(((tile_d0 >> 16) & 0xFFFFu)         // tensor_dim0[31:16]
                | ((tile_d1 & 0xFFFFu) << 16));       // tensor_dim1[15:0]
    g1[3] = (int)(((tile_d1 >> 16) & 0xFFFFu)         // tensor_dim1[31:16]
                | ((tile_d0 & 0xFFFFu) << 16));       // tile_dim0
    g1[4] = (int)(tile_d1 & 0xFFFFu);                 // tile_dim1 (tile_dim2=0)
    g1[5] = (int)stride0;                             // tensor_dim0_stride[31:0]
    g1[6] = 0;                                        // stride0 hi / stride1 lo
    g1[7] = 0;
    v4i gz4  = {0, 0, 0, 0};
    v8i gz8  = {0, 0, 0, 0, 0, 0, 0, 0};
    // 6-arg variant (clang-23 / therock-10.0 headers)
    __builtin_amdgcn_tensor_load_to_lds(g0, g1, gz4, gz4, gz8, 0);
}

__device__ __forceinline__ void wait_tensor0() {
    __builtin_amdgcn_s_wait_tensorcnt(0);
}

// ---------------------------------------------------------------------------
// Stage 1: build the 20 step matrices M_i = N_i @ B_i on device, in double
// precision (reference builds in float64 then casts to float32).
// Row j of M_i equals row src(j) of the bicubic matrix B_i (s x 256).
// ---------------------------------------------------------------------------
__global__ void build_M_kernel(float* __restrict__ Ms) {
    const int i = blockIdx.x;     // step index, 0..19
    const int j = threadIdx.x;    // output row, 0..255
    const int s = IMG - i;

    double xs = ((double)j + 0.5) * (double)s / (double)IMG;
    int r = (int)floor(xs);
    if (r < 0) r = 0;
    if (r > s - 1) r = s - 1;

    const double Ac = -0.75;
    double x  = ((double)r + 0.5) * ((double)IMG / (double)s) - 0.5;
    double f0 = floor(x);
    int   i0  = (int)f0;
    double tt = x - f0;

    double tp1 = tt + 1.0, omt = 1.0 - tt, tm2 = 2.0 - tt;
    double w[4];
    w[0] = ((Ac * tp1 - 5.0 * Ac) * tp1 + 8.0 * Ac) * tp1 - 4.0 * Ac; // _cc2(t+1)
    w[1] = ((Ac + 2.0) * tt  - (Ac + 3.0)) * tt  * tt  + 1.0;         // _cc1(t)
    w[2] = ((Ac + 2.0) * omt - (Ac + 3.0)) * omt * omt + 1.0;         // _cc1(1-t)
    w[3] = ((Ac * tm2 - 5.0 * Ac) * tm2 + 8.0 * Ac) * tm2 - 4.0 * Ac; // _cc2(2-t)

    int idx[4];
    #pragma unroll
    for (int k = 0; k < 4; ++k) {
        int c = i0 - 1 + k;
        if (c < 0) c = 0;
        if (c > IMG - 1) c = IMG - 1;
        idx[k] = c;
    }

    float* row = Ms + (size_t)i * MAT_EL + (size_t)j * IMG;
    for (int c = 0; c < IMG; ++c) row[c] = 0.0f;

    #pragma unroll
    for (int k = 0; k < 4; ++k) {
        bool first = true;
        double sum = 0.0;
        #pragma unroll
        for (int m = 0; m < 4; ++m) {
            if (idx[m] == idx[k]) {
                if (m < k) first = false;
                sum += w[m];
            }
        }
        if (first) row[idx[k]] = (float)sum;
    }
}

// P_0 = identity
__global__ void init_P0_kernel(float* __restrict__ P) {
    int idx = blockIdx.x * blockDim.x + threadIdx.x;   // 0..65535
    int r = idx >> 8, c = idx & 255;
    P[idx] = (r == c) ? 1.0f : 0.0f;
}

// ---------------------------------------------------------------------------
// 256x256 GEMM core: C = A * B (transB=false) or C = A * B^T.
// Block: 256 threads = 8 waves; block tile 128(M) x 64(N); wave grid 4x2;
// each wave owns 32x32 = 2x2 V_WMMA_F32_16X16X4_F32 accumulators.
// K pipelined in chunks of 16 through double-buffered LDS.
// Staging: useTdm ? one TENSOR_LOAD_TO_LDS per tile issued by wave 0
//                 : per-lane GLOBAL_LOAD_ASYNC_TO_LDS_B128.
// One barrier per chunk; next chunk's DMA overlaps current WMMAs.
// ---------------------------------------------------------------------------
__device__ __forceinline__ void gemm256_core(const float* __restrict__ A,
                                             const float* __restrict__ B,
                                             float* __restrict__ C,
                                             bool transB, bool useTdm)
{
    __shared__ float lds[2 * BUF_F];     // 30 KB, double-buffered A+B tiles

    const int tid   = threadIdx.x;
    const int lane  = tid & 31;
    const int wave  = tid >> 5;
    const int l     = lane & 15;
    const int half  = lane >> 4;
    const int waveM = wave & 3;          // 0..3 -> 32-row strip
    const int waveN = wave >> 2;         // 0..1 -> 32-col strip
    const int rowBase = blockIdx.y * 128;
    const int colBase = blockIdx.x * 64;

    const unsigned base = lds_off_of(lds);

    // per-thread staging coordinates for the async (non-TDM) path
    const int aRow0 = (tid + 0)   >> 2, aSeg0 = (tid + 0)   & 3;
    const int aRow1 = (tid + 256) >> 2, aSeg1 = (tid + 256) & 3;
    const int bRowNN = tid >> 4, bSegNN = tid & 15;      // Bs[k][n]
    const int bRowNT = tid >> 2, bSegNT = tid & 3;       // Bts[n][k]

    // stage K-chunk `kb` into buffer `buf`
    auto stage = [&](int kb, int buf) {
        const unsigned abase = base + (unsigned)(buf * BUF_F) * 4u;
        const unsigned bbase = abase + (unsigned)A_TILE_F * 4u;
        if (useTdm) {
            if (wave == 0) {
                // A tile: 128 rows x 16 elems, LDS stride 20 (pad 16dw + 4dw)
                tdm_load_2d(abase, A + (size_t)rowBase * IMG + kb,
                            KC, 128, IMG, /*pad_iv=*/3, /*pad_amt=*/3);
                if (!transB) {
                    // B tile: 16 rows x 64 elems, stride 72 (pad 64dw + 8dw)
                    tdm_load_2d(bbase, B + (size_t)kb * IMG + colBase,
                                64, KC, IMG, /*pad_iv=*/5, /*pad_amt=*/7);
                } else {
                    // Bt tile: 64 rows x 16 elems, stride 20 (pad 16dw + 4dw)
                    tdm_load_2d(bbase, B + (size_t)colBase * IMG + kb,
                                KC, 64, IMG, /*pad_iv=*/3, /*pad_amt=*/3);
                }
            }
        } else {
            async_copy16(abase + (unsigned)(aRow0 * A_STRIDE) * 4u + aSeg0 * 16u,
                         A + (size_t)(rowBase + aRow0) * IMG + kb + aSeg0 * 4);
            async_copy16(abase + (unsigned)(aRow1 * A_STRIDE) * 4u + aSeg1 * 16u,
                         A + (size_t)(rowBase + aRow1) * IMG + kb + aSeg1 * 4);
            if (!transB) {
                async_copy16(bbase + (unsigned)(bRowNN * BNN_STRIDE) * 4u + bSegNN * 16u,
                             B + (size_t)(kb + bRowNN) * IMG + colBase + bSegNN * 4);
            } else {
                async_copy16(bbase + (unsigned)(bRowNT * BNT_STRIDE) * 4u + bSegNT * 16u,
                             B + (size_t)(colBase + bRowNT) * IMG + kb + bSegNT * 4);
            }
        }
    };

    v8f acc[2][2] = {};

    stage(0, 0);                                   // prologue

    const int NKB = IMG / KC;                      // 16 chunks
    for (int it = 0; it < NKB; ++it) {
        const int buf = it & 1;
        // producer(s) wait for their DMA into `buf`, then one barrier
        // releases all consumers; the same barrier proves everyone finished
        // reading the other buffer, so it may be restaged immediately.
        if (useTdm) {
            if (wave == 0) wait_tensor0();
        } else {
            wait_async0();
        }
        __syncthreads();
        if (it + 1 < NKB) stage((it + 1) * KC, buf ^ 1);   // overlap next DMA

        const float* Ab = lds + buf * BUF_F;
        const float* Bb = Ab + A_TILE_F;

        #pragma unroll
        for (int kk = 0; kk < KC; kk += 4) {
            v2f afr[2], bfr[2];
            #pragma unroll
            for (int tm = 0; tm < 2; ++tm)
                afr[tm] = *(const v2f*)&Ab[(waveM * 32 + tm * 16 + l) * A_STRIDE
                                           + kk + 2 * half];
            #pragma unroll
            for (int tn = 0; tn < 2; ++tn) {
                const int c0 = waveN * 32 + tn * 16 + l;
                if (!transB) {
                    bfr[tn].x = Bb[(kk + 2 * half) * BNN_STRIDE + c0];
                    bfr[tn].y = Bb[(kk + 2 * half + 1) * BNN_STRIDE + c0];
                } else {
                    bfr[tn] = *(const v2f*)&Bb[c0 * BNT_STRIDE + kk + 2 * half];
                }
            }
            #pragma unroll
            for (int tm = 0; tm < 2; ++tm)
                #pragma unroll
                for (int tn = 0; tn < 2; ++tn)
                    acc[tm][tn] = __builtin_amdgcn_wmma_f32_16x16x4_f32(
                        /*neg_a=*/false, afr[tm],
                        /*neg_b=*/false, bfr[tn],
                        /*c_mod=*/(short)0, acc[tm][tn],
                        /*reuse_a=*/false, /*reuse_b=*/false);
        }
    }

    // store C per documented 16x16 f32 C/D layout
    #pragma unroll
    for (int tm = 0; tm < 2; ++tm)
        #pragma unroll
        for (int tn = 0; tn < 2; ++tn)
            #pragma unroll
            for (int e = 0; e < 8; ++e) {
                int rr = rowBase + waveM * 32 + tm * 16 + e + 8 * half;
                int cc = colBase + waveN * 32 + tn * 16 + l;
                C[(size_t)rr * IMG + cc] = acc[tm][tn][e];
            }
}

// prefix-product step: C = A * B (tiny sequential GEMMs, per-lane async path)
__global__ void gemm_nn_kernel(const float* __restrict__ A,
                               const float* __restrict__ B,
                               float* __restrict__ C) {
    gemm256_core(A, B, C, false, false);
}

// pass 1: tmp[z] = P_{t[b]} * x0[z]     (z = b*CH + c)  — TDM staging
__global__ void pass1_kernel(const float* __restrict__ P,
                             const int* __restrict__ t,
                             const float* __restrict__ x0,
                             float* __restrict__ tmp) {
    const int z  = blockIdx.z;
    const int b  = z / CH;
    const int ts = t[b];
    gemm256_core(P  + (size_t)ts * MAT_EL,
                 x0 + (size_t)z  * MAT_EL,
                 tmp + (size_t)z * MAT_EL, false, true);
}

// pass 2: out[z] = tmp[z] * P_{t[b]}^T  — TDM staging
__global__ void pass2_kernel(const float* __restrict__ tmp,
                             const int* __restrict__ t,
                             const float* __restrict__ P,
                             float* __restrict__ out) {
    const int z  = blockIdx.z;
    const int b  = z / CH;
    const int ts = t[b];
    gemm256_core(tmp + (size_t)z  * MAT_EL,
                 P   + (size_t)ts * MAT_EL,
                 out + (size_t)z  * MAT_EL, true, true);
}

// ---------------------------------------------------------------------------
// Workspace layout (floats):
//   Ms  : [0,                20*65536)   5 MB  — step matrices
//   P   : [20*65536,         40*65536)   5 MB  — prefix products P_0..P_19
//   tmp : [40*65536,        232*65536)  48 MB  — pass-1 intermediates
// Total ~61 MB of d_ws.
// ---------------------------------------------------------------------------
extern "C" void kernel_launch(void* const* d_in, const int* in_sizes, int n_in,
                              void* d_out, int out_size, void* d_ws, size_t ws_size,
                              hipStream_t stream) {
    const float* x0 = (const float*)d_in[0];
    const int*   t  = (const int*)d_in[1];
    float* out = (float*)d_out;
    float* ws  = (float*)d_ws;

    float* Ms  = ws;
    float* P   = ws + (size_t)TSTEPS * MAT_EL;
    float* tmp = ws + (size_t)2 * TSTEPS * MAT_EL;

    // Stage 1: step matrices + identity P_0
    build_M_kernel<<<TSTEPS, IMG, 0, stream>>>(Ms);
    init_P0_kernel<<<MAT_EL / 256, 256, 0, stream>>>(P);

    // Stage 2: prefix products P_s = M_{s-1} * P_{s-1}  (sequential, tiny)
    dim3 g1(IMG / 64, IMG / 128, 1);   // (4, 2)
    for (int s = 1; s < TSTEPS; ++s) {
        gemm_nn_kernel<<<g1, 256, 0, stream>>>(Ms + (size_t)(s - 1) * MAT_EL,
                                               P  + (size_t)(s - 1) * MAT_EL,
                                               P  + (size_t)s * MAT_EL);
    }

    // Stage 3/4: batched out = P_t * x0 * P_t^T over all 192 images
    dim3 gb(IMG / 64, IMG / 128, BATCH * CH);
    pass1_kernel<<<gb, 256, 0, stream>>>(P, t, x0, tmp);
    pass2_kernel<<<gb, 256, 0, stream>>>(tmp, t, P, out);
}